// MolecularGAT0_103079215297
// MI455X (gfx1250) — compile-verified
//
#include <hip/hip_runtime.h>

// MI455X / gfx1250: wave32, WMMA f32 16x16x4 for exact fp32 GEMM phases.
// One workgroup per batch graph; all per-graph state held in LDS so HBM
// traffic equals the compulsory 207 MB (~9us at 23.3 TB/s).

typedef float v2f __attribute__((ext_vector_type(2)));
typedef float v8f __attribute__((ext_vector_type(8)));

#define BATCH 512
#define NN    128
#define AD    75
#define KPAD  80      // 75 padded to multiple of 16 (and of 4 for K-steps)
#define LDX   81      // row stride for atoms/xl/W tiles (bank-conflict-free A loads)
#define LDA   136     // row stride for attn matrix (half-wave rows hit disjoint banks)

// LDS layout (floats)
#define OFF_ATOMS  0                    // 128*81 = 10368
#define OFF_W      10368                // 80*81  = 6480
#define OFF_XL     16848                // 128*81 = 10368
#define OFF_ATTN   27216                // 128*136= 17408
#define OFF_ASRC   44624                // 128
#define OFF_ADST   44752                // 128
#define OFF_VSRC   44880                // 80
#define OFF_VDST   44960                // 80
#define OFF_BIAS   45040                // 80
#define OFF_WE     45120                // 4 (+pad)
#define SMEM_FLOATS 45128

__global__ void __launch_bounds__(256, 1)
molgat_wmma_kernel(const float* __restrict__ atoms,     // [B,128,75]
                   const float* __restrict__ adjs,      // [B,128,128]
                   const float* __restrict__ edges,     // [B,128,128,4]
                   const float* __restrict__ W,         // [75,75]
                   const float* __restrict__ We,        // [4,75]
                   const float* __restrict__ att_src,   // [75]
                   const float* __restrict__ att_dst,   // [75]
                   const float* __restrict__ att_edge,  // [75]
                   const float* __restrict__ bias,      // [75]
                   float* __restrict__ out)             // [B,128,75]
{
    extern __shared__ float smem[];
    float* sAtoms = smem + OFF_ATOMS;
    float* sW     = smem + OFF_W;
    float* sXl    = smem + OFF_XL;
    float* sAttn  = smem + OFF_ATTN;
    float* sAsrc  = smem + OFF_ASRC;
    float* sAdst  = smem + OFF_ADST;
    float* sVsrc  = smem + OFF_VSRC;
    float* sVdst  = smem + OFF_VDST;
    float* sBias  = smem + OFF_BIAS;
    float* sWe    = smem + OFF_WE;

    const int tid  = threadIdx.x;
    const int lane = tid & 31;
    const int wave = tid >> 5;         // 8 waves
    const int b    = blockIdx.x;
    const int hi   = lane >> 4;        // half-wave select
    const int mn   = lane & 15;        // M (A/C rows) or N (B/C cols) index
    const int koff = hi * 2;           // K offset per 16x16x4 fragment layout
    const int mBase = wave * 16;       // each wave owns a 16-row M block

    // ---------------- Phase 0: stage operands into LDS ----------------
    for (int idx = tid; idx < NN * LDX; idx += 256) {
        int n = idx / LDX, c = idx - n * LDX;
        sAtoms[idx] = (c < AD) ? atoms[(size_t)b * NN * AD + n * AD + c] : 0.f;
    }
    for (int idx = tid; idx < KPAD * LDX; idx += 256) {
        int r = idx / LDX, c = idx - r * LDX;
        sW[idx] = (r < AD && c < AD) ? W[r * AD + c] : 0.f;
    }
    if (tid < KPAD) {
        sVsrc[tid] = (tid < AD) ? att_src[tid] : 0.f;
        sVdst[tid] = (tid < AD) ? att_dst[tid] : 0.f;
        sBias[tid] = (tid < AD) ? bias[tid]    : 0.f;
    }
    if (tid < 4) {  // fold lin_edge with att_edge: w_edge[e] = <We[e,:], att_edge>
        float s = 0.f;
        for (int o = 0; o < AD; ++o) s += We[tid * AD + o] * att_edge[o];
        sWe[tid] = s;
    }
    __syncthreads();

    // ---------------- Phase 1: xl = atoms @ W  (WMMA f32 16x16x4) ------
    {
        v8f acc[5] = {};
        for (int kk = 0; kk < KPAD; kk += 4) {
            const int k0 = kk + koff;
            v2f a;
            a.x = sAtoms[(mBase + mn) * LDX + k0];
            a.y = sAtoms[(mBase + mn) * LDX + k0 + 1];
#pragma unroll
            for (int s = 0; s < 5; ++s) {
                v2f bb;
                bb.x = sW[k0 * LDX + s * 16 + mn];
                bb.y = sW[(k0 + 1) * LDX + s * 16 + mn];
                acc[s] = __builtin_amdgcn_wmma_f32_16x16x4_f32(
                    false, a, false, bb, (short)0, acc[s], false, false);
            }
        }
#pragma unroll
        for (int s = 0; s < 5; ++s)
#pragma unroll
            for (int r = 0; r < 8; ++r)
                sXl[(mBase + 8 * hi + r) * LDX + s * 16 + mn] = acc[s][r];
    }
    __syncthreads();

    // ---------------- Phase 2a: per-node attention terms ---------------
    if (tid < NN) {
        float s = 0.f, d = 0.f;
        for (int o = 0; o < AD; ++o) {
            float x = sXl[tid * LDX + o];
            s += x * sVsrc[o];
            d += x * sVdst[o];
        }
        sAsrc[tid] = s;
        sAdst[tid] = d;
    }
    __syncthreads();

    // ---------------- Phase 2b: alpha + leaky-relu + mask (streaming) --
    {
        const float4* edg = (const float4*)(edges + (size_t)b * NN * NN * 4);
        const float*  adj = adjs + (size_t)b * NN * NN;
        const float w0 = sWe[0], w1 = sWe[1], w2 = sWe[2], w3 = sWe[3];
        for (int idx = tid; idx < NN * NN; idx += 256) {
            int j = idx >> 7, i = idx & 127;       // j = source, i = target
            float4 e = edg[idx];
            float a = sAsrc[j] + sAdst[i] + (e.x * w0 + e.y * w1 + e.z * w2 + e.w * w3);
            a = (a > 0.f) ? a : 0.2f * a;          // leaky relu, slope 0.2
            a = (adj[idx] > 0.5f) ? a : -1e9f;     // mask
            sAttn[j * LDA + i] = a;
        }
    }
    __syncthreads();

    // ---------------- Phase 3: masked softmax over sources j -----------
    if (tid < NN) {
        const int i = tid;
        float m = -1e30f;
        for (int j = 0; j < NN; ++j) m = fmaxf(m, sAttn[j * LDA + i]);
        float sum = 0.f;
        for (int j = 0; j < NN; ++j) {
            float a = sAttn[j * LDA + i];
            float p = (a > -5e8f) ? __expf(a - m) : 0.f;  // masked -> 0
            sAttn[j * LDA + i] = p;
            sum += p;
        }
        float inv = 1.f / (sum + 1e-16f);
        for (int j = 0; j < NN; ++j) sAttn[j * LDA + i] *= inv;
    }
    __syncthreads();

    // ---------------- Phase 4: out = attn^T @ xl  (WMMA f32 16x16x4) ---
    {
        v8f acc[5] = {};
        for (int kk = 0; kk < NN; kk += 4) {
            const int k0 = kk + koff;
            v2f a;   // A[m][k] = attn[k][m]  (transpose read, lane-consecutive)
            a.x = sAttn[k0 * LDA + mBase + mn];
            a.y = sAttn[(k0 + 1) * LDA + mBase + mn];
#pragma unroll
            for (int s = 0; s < 5; ++s) {
                v2f bb;
                bb.x = sXl[k0 * LDX + s * 16 + mn];
                bb.y = sXl[(k0 + 1) * LDX + s * 16 + mn];
                acc[s] = __builtin_amdgcn_wmma_f32_16x16x4_f32(
                    false, a, false, bb, (short)0, acc[s], false, false);
            }
        }
        // epilogue: + bias, write only o < 75, coalesced 16-wide row segments
#pragma unroll
        for (int s = 0; s < 5; ++s) {
            const int o = s * 16 + mn;
            if (o < AD) {
                const float bv = sBias[o];
#pragma unroll
                for (int r = 0; r < 8; ++r) {
                    const int i = mBase + 8 * hi + r;
                    out[(size_t)b * NN * AD + i * AD + o] = acc[s][r] + bv;
                }
            }
        }
    }
}

extern "C" void kernel_launch(void* const* d_in, const int* in_sizes, int n_in,
                              void* d_out, int out_size, void* d_ws, size_t ws_size,
                              hipStream_t stream) {
    const float* atoms    = (const float*)d_in[0];
    const float* adjs     = (const float*)d_in[1];
    const float* edges    = (const float*)d_in[2];
    const float* W        = (const float*)d_in[3];
    const float* We       = (const float*)d_in[4];
    const float* att_src  = (const float*)d_in[5];
    const float* att_dst  = (const float*)d_in[6];
    const float* att_edge = (const float*)d_in[7];
    const float* bias     = (const float*)d_in[8];
    float* out = (float*)d_out;

    constexpr size_t smemBytes = (size_t)SMEM_FLOATS * sizeof(float);
    // Allow >64KB dynamic LDS (CDNA5 WGP has 320KB).
    hipFuncSetAttribute((const void*)molgat_wmma_kernel,
                        hipFuncAttributeMaxDynamicSharedMemorySize, (int)smemBytes);

    molgat_wmma_kernel<<<dim3(BATCH), dim3(256), smemBytes, stream>>>(
        atoms, adjs, edges, W, We, att_src, att_dst, att_edge, bias, out);
}